// ScaledDotProductAttention_5188320493945
// MI455X (gfx1250) — compile-verified
//
#include <hip/hip_runtime.h>
#include <hip/hip_bf16.h>

typedef __attribute__((ext_vector_type(2))) float v2f;
typedef __attribute__((ext_vector_type(4))) float v4f;
typedef __attribute__((ext_vector_type(8))) float v8f;

constexpr int B = 4, H = 16, S = 2048, D = 64;
constexpr int QT = 16;           // queries per workgroup
constexpr int NT = S / QT;       // 128 key tiles of 16
constexpr int SROW = S + 4;      // padded LDS row stride (dwords) -> bank shift of 4/row
constexpr int NWAVE = 8;

__global__ __launch_bounds__(256) void attn_fused_kernel(
    const float* __restrict__ Q, const float* __restrict__ K,
    const float* __restrict__ V, const int* __restrict__ mask,
    const float* __restrict__ bias, float* __restrict__ out,
    float* __restrict__ attn)
{
  // 16 x 2048 fp32 score row (+pad)   = 131,328 B
  // 8-wave partial O tiles (16x64)    =  32,768 B
  // reduction scratch                 ~   1.2 KB   (CDNA5: 320 KB LDS / WGP)
  __shared__ float s_scores[QT * SROW];
  __shared__ float s_obuf[NWAVE * QT * D];
  __shared__ float s_red[256];
  __shared__ float s_rowmax[QT];
  __shared__ float s_rowinv[QT];

  const int tid  = threadIdx.x;
  const int wave = tid >> 5;
  const int lane = tid & 31;
  const int l16  = lane & 15;    // M / N index within fragment
  const int lhi  = lane >> 4;    // 0 or 1: selects K-half / M-half

  const int tileid = blockIdx.x;
  const int qt = tileid % (S / QT);
  const int bh = tileid / (S / QT);
  const int h  = bh % H;
  const int b  = bh / H;
  const int q0 = qt * QT;

  const float scale = 0.125f;    // 1/sqrt(64)

  // ---------- load Q A-fragments: 16x64 fp32 = 16 WMMA K-steps of 16x4 ----------
  // A layout (32-bit, 16x4): lane holds A[M = lane%16][K = 2*(lane/16) + v]
  v2f qa[16];
  {
    const float* qrow = Q + ((size_t)bh * S + q0 + l16) * D + 2 * lhi;
#pragma unroll
    for (int kk = 0; kk < 16; ++kk)
      qa[kk] = *reinterpret_cast<const v2f*>(qrow + kk * 4);
  }

  // ---------- phase 1: scores = Q K^T * scale + bias, key-pad mask -> LDS ----------
  for (int jt = wave; jt < NT; jt += NWAVE) {
    const int k0 = jt * QT;
    v8f c = {};
    // B layout (32-bit, 4x16): lane holds B[K = 2*(lane/16)+v][N = lane%16] = Kmat[k0+N][K]
    const float* krow = K + ((size_t)bh * S + k0 + l16) * D + 2 * lhi;
#pragma unroll
    for (int kk = 0; kk < 16; ++kk) {
      v2f bf = *reinterpret_cast<const v2f*>(krow + kk * 4);
      c = __builtin_amdgcn_wmma_f32_16x16x4_f32(false, qa[kk], false, bf,
                                                (short)0, c, false, false);
    }
    // C layout: VGPR r -> row = r + 8*(lane/16), col = lane%16
#pragma unroll
    for (int r = 0; r < 8; ++r) {
      const int row  = r + 8 * lhi;
      const int kcol = k0 + l16;
      float sc = c[r] * scale + bias[((size_t)h * S + (q0 + row)) * S + kcol];
      if (mask[b * S + kcol] == 0) sc = -1e9f;
      s_scores[row * SROW + kcol] = sc;
    }
  }
  __syncthreads();

  // ---------- phase 2: row max, then exp-sum (16 threads/row, stride-16 cols) ----------
  const int row = tid >> 4;
  const int seg = tid & 15;
  {
    float m = -3.4e38f;
    for (int j = 0; j < S / 16; ++j)
      m = fmaxf(m, s_scores[row * SROW + seg + 16 * j]);
    s_red[tid] = m;
  }
  __syncthreads();
  if (tid < QT) {
    float m = s_red[tid * 16];
#pragma unroll
    for (int j = 1; j < 16; ++j) m = fmaxf(m, s_red[tid * 16 + j]);
    s_rowmax[tid] = m;
  }
  __syncthreads();
  {
    const float mr = s_rowmax[row];
    float s = 0.f;
    for (int j = 0; j < S / 16; ++j)
      s += __expf(s_scores[row * SROW + seg + 16 * j] - mr);
    s_red[tid] = s;
  }
  __syncthreads();
  if (tid < QT) {
    float s = 0.f;
#pragma unroll
    for (int j = 0; j < 16; ++j) s += s_red[tid * 16 + j];
    s_rowinv[tid] = 1.f / s;
  }
  __syncthreads();

  // ---------- phase 3: normalize; write attn (coalesced b128); keep P in LDS ----------
  for (int r = 0; r < QT; ++r) {
    const float  mr   = s_rowmax[r];
    const float  inv  = s_rowinv[r];
    const size_t arow = ((size_t)(b * H + h) * S + (q0 + r)) * S;
    for (int i = tid; i < S / 4; i += 256) {
      v4f x = *reinterpret_cast<v4f*>(&s_scores[r * SROW + i * 4]);
      v4f p;
      p.x = __expf(x.x - mr) * inv;
      p.y = __expf(x.y - mr) * inv;
      p.z = __expf(x.z - mr) * inv;
      p.w = __expf(x.w - mr) * inv;
      *reinterpret_cast<v4f*>(&s_scores[r * SROW + i * 4]) = p;
      *reinterpret_cast<v4f*>(attn + arow + i * 4)         = p;
    }
  }
  __syncthreads();

  // ---------- phase 4: O = P(16x2048) * V(2048x64), fp32 WMMA, 4 n-tiles ----------
  v8f acc[4] = {v8f{}, v8f{}, v8f{}, v8f{}};
  for (int jt = wave; jt < NT; jt += NWAVE) {
#pragma unroll
    for (int kk = 0; kk < 4; ++kk) {
      const int kbase = jt * QT + kk * 4 + 2 * lhi;   // dword-even -> ds_load_b64 aligned
      v2f pa = *reinterpret_cast<v2f*>(&s_scores[l16 * SROW + kbase]);
      const float* v0 = V + ((size_t)bh * S + kbase) * D + l16;
#pragma unroll
      for (int nt = 0; nt < 4; ++nt) {
        v2f bf;
        bf.x = v0[nt * 16];        // B[K=kbase  ][N = nt*16 + lane%16]
        bf.y = v0[D + nt * 16];    // B[K=kbase+1][N]
        acc[nt] = __builtin_amdgcn_wmma_f32_16x16x4_f32(false, pa, false, bf,
                                                        (short)0, acc[nt], false, false);
      }
    }
  }
  // partial O tiles -> LDS
#pragma unroll
  for (int nt = 0; nt < 4; ++nt) {
#pragma unroll
    for (int r = 0; r < 8; ++r) {
      const int orow = r + 8 * lhi;
      const int ocol = nt * 16 + l16;
      s_obuf[(wave * QT + orow) * D + ocol] = acc[nt][r];
    }
  }
  __syncthreads();
  // cross-wave reduction + coalesced output store
  for (int e = tid; e < QT * D; e += 256) {
    float sum = 0.f;
#pragma unroll
    for (int w = 0; w < NWAVE; ++w) sum += s_obuf[w * QT * D + e];
    out[((size_t)(b * H + h) * S + (q0 + (e >> 6))) * D + (e & 63)] = sum;
  }
}

extern "C" void kernel_launch(void* const* d_in, const int* in_sizes, int n_in,
                              void* d_out, int out_size, void* d_ws, size_t ws_size,
                              hipStream_t stream) {
  (void)in_sizes; (void)n_in; (void)out_size; (void)d_ws; (void)ws_size;
  const float* Q    = (const float*)d_in[0];
  const float* K    = (const float*)d_in[1];
  const float* V    = (const float*)d_in[2];
  const int*   mask = (const int*)d_in[3];
  const float* bias = (const float*)d_in[4];
  float* out  = (float*)d_out;
  float* attn = out + (size_t)B * H * S * D;   // tuple: (output, attn) flat-concatenated

  dim3 grid(B * H * (S / QT));                 // 8192 workgroups, one 16-query row each
  attn_fused_kernel<<<grid, 256, 0, stream>>>(Q, K, V, mask, bias, out, attn);
}